// CGMPModel_28741921145488
// MI455X (gfx1250) — compile-verified
//
#include <hip/hip_runtime.h>
#include <hip/hip_bf16.h>

#define NN      4096      // nodes
#define NE      131072    // edges
#define NCIN    8
#define NCC     16        // NC
#define WPB     8         // waves per block in streaming WMMA kernels
#define WPC     4         // waves per block in cg kernel (LDS-limited)

typedef __attribute__((ext_vector_type(2))) float v2f;
typedef __attribute__((ext_vector_type(8))) float v8f;
typedef __attribute__((ext_vector_type(2))) int   v2i;

// ---------------- CG coefficient tables -----------------------------------
// pair types: 0:(0,0,0) 1:(1,1,0) 2:(0,1,1) 3:(1,0,1) 4:(1,1,1) 5:(1,1,2) 6:(2,2,0)
struct CGTerm { int a; int b; float c; };

__constant__ int PT_LA[7] = {0,1,0,1,1,1,2};
__constant__ int PT_LB[7] = {0,1,1,0,1,1,2};

__constant__ CGTerm CG_TERMS[30] = {
  /* pt0 m0 */ {0,0, 1.0f},
  /* pt1 m0 */ {0,2, 0.57735027f},{1,1,-0.57735027f},{2,0, 0.57735027f},
  /* pt2 m0..2 */ {0,0,1.0f},{0,1,1.0f},{0,2,1.0f},
  /* pt3 m0..2 */ {0,0,1.0f},{1,0,1.0f},{2,0,1.0f},
  /* pt4 m0 */ {1,0, 0.70710678f},{0,1,-0.70710678f},
  /* pt4 m1 */ {2,0, 0.70710678f},{0,2,-0.70710678f},
  /* pt4 m2 */ {2,1, 0.70710678f},{1,2,-0.70710678f},
  /* pt5 m0 */ {0,0,1.0f},
  /* pt5 m1 */ {0,1,0.70710678f},{1,0,0.70710678f},
  /* pt5 m2 */ {0,2,0.40824829f},{1,1,0.81649658f},{2,0,0.40824829f},
  /* pt5 m3 */ {1,2,0.70710678f},{2,1,0.70710678f},
  /* pt5 m4 */ {2,2,1.0f},
  /* pt6 m0 */ {0,4,0.4472136f},{1,3,-0.4472136f},{2,2,0.4472136f},
               {3,1,-0.4472136f},{4,0,0.4472136f}
};
__constant__ int CG_OFF[7][5] = {
  {0,0,0,0,0},{1,0,0,0,0},{4,5,6,0,0},{7,8,9,0,0},
  {10,12,14,0,0},{16,17,19,22,24},{25,0,0,0,0}
};
__constant__ int CG_CNT[7][5] = {
  {1,0,0,0,0},{3,0,0,0,0},{1,1,1,0,0},{1,1,1,0,0},
  {2,2,2,0,0},{1,2,3,2,1},{5,0,0,0,0}
};

// ---------------- async global -> LDS staging helper ------------------------
__device__ __forceinline__ void stage_f2(const float2* g, float2* l) {
#if defined(__HIP_DEVICE_COMPILE__) && \
    __has_builtin(__builtin_amdgcn_global_load_async_to_lds_b64)
  __builtin_amdgcn_global_load_async_to_lds_b64(
      (__attribute__((address_space(1))) v2i*)(g),
      (__attribute__((address_space(3))) v2i*)(l), 0, 0);
#else
  *l = *g;
#endif
}
__device__ __forceinline__ void stage_wait() {
#if defined(__HIP_DEVICE_COMPILE__) && \
    __has_builtin(__builtin_amdgcn_s_wait_asynccnt)
  __builtin_amdgcn_s_wait_asynccnt(0);
#endif
}

// ---------------- utility kernels -----------------------------------------
__global__ __launch_bounds__(256) void zero_kernel(float* p, size_t n) {
  size_t i = (size_t)blockIdx.x * blockDim.x + threadIdx.x;
  if (i < n) p[i] = 0.0f;
}

// initial_layer edge scatter: x1[dst,m, k*8+c] += sph[m]*g[k]*x[src,c]
__global__ __launch_bounds__(256) void init_edge_kernel(
    const float* __restrict__ pos, const float* __restrict__ x,
    const int* __restrict__ eidx, float2* __restrict__ x1) {
  int e = blockIdx.x * blockDim.x + threadIdx.x;
  if (e >= NE) return;
  int src = eidx[e];
  int dst = eidx[NE + e];
  float rx = pos[src*3+0] - pos[dst*3+0];
  float ry = pos[src*3+1] - pos[dst*3+1];
  float rz = pos[src*3+2] - pos[dst*3+2];
  float r  = sqrtf(rx*rx + ry*ry + rz*rz + 1e-12f);
  float iv = 1.0f / (r + 1e-8f);
  float ux = rx*iv, uy = ry*iv, uz = rz*iv;
  const float is2 = 0.70710678f;
  float2 sph[3];
  sph[0] = make_float2( ux*is2, -uy*is2);
  sph[1] = make_float2( uz,      0.0f  );
  sph[2] = make_float2(-ux*is2, -uy*is2);
  float g[4];
  #pragma unroll
  for (int k = 0; k < 4; ++k) { float d = r - (float)(k+1); g[k] = expf(-d*d); }
  for (int m = 0; m < 3; ++m) {
    float2 s = sph[m];
    for (int k = 0; k < 4; ++k) {
      for (int c = 0; c < NCIN; ++c) {
        float w = g[k] * x[src*NCIN + c];
        float2* o = &x1[(size_t)(dst*3 + m)*32 + k*NCIN + c];
        atomicAdd(&o->x, s.x * w);
        atomicAdd(&o->y, s.y * w);
      }
    }
  }
}

// mp_layer scatter: y[dst,m,e2*nc+c] += eattr[e,e2]*x[src,m,c]
__global__ __launch_bounds__(256) void mp_kernel(
    const float2* __restrict__ x, const int* __restrict__ eidx,
    const float* __restrict__ eatt, float2* __restrict__ y,
    int nm, int nc) {
  int e = blockIdx.x * blockDim.x + threadIdx.x;
  if (e >= NE) return;
  int src = eidx[e];
  int dst = eidx[NE + e];
  float ev0 = eatt[e*2+0], ev1 = eatt[e*2+1];
  for (int m = 0; m < nm; ++m) {
    for (int c = 0; c < nc; ++c) {
      float2 v = x[(size_t)(src*nm + m)*nc + c];
      float2* o0 = &y[(size_t)(dst*nm + m)*(2*nc) + c];
      float2* o1 = &y[(size_t)(dst*nm + m)*(2*nc) + nc + c];
      atomicAdd(&o0->x, ev0*v.x); atomicAdd(&o0->y, ev0*v.y);
      atomicAdd(&o1->x, ev1*v.x); atomicAdd(&o1->y, ev1*v.y);
    }
  }
}

// ---------------- WMMA GEMM kernels (V_WMMA_F32_16X16X4_F32) --------------
// so3_linear on real input: out[row,d].re += sum_k X[row,k]*W[k,d]
__global__ __launch_bounds__(32*WPB) void lin_real_kernel(
    const float* __restrict__ X, const float* __restrict__ W,
    float2* __restrict__ out, int K) {
  int lane = threadIdx.x & 31, wave = threadIdx.x >> 5;
  int lo = lane & 15, hi = lane >> 4;
  int row0 = (blockIdx.x * WPB + wave) * 16;
  v8f acc = {};
  for (int k0 = 0; k0 < K; k0 += 4) {
    v2f a, b;
    #pragma unroll
    for (int j = 0; j < 2; ++j) {
      int kk = k0 + 2*hi + j;
      a[j] = X[(size_t)(row0 + lo)*K + kk];   // A: M=lane&15, K split by half-wave
      b[j] = W[kk*NCC + lo];                  // B: N=lane&15, same K split
    }
    acc = __builtin_amdgcn_wmma_f32_16x16x4_f32(false, a, false, b, (short)0, acc,
                                                false, false);
  }
  #pragma unroll
  for (int i = 0; i < 8; ++i) {              // D: N=lane&15, M=i+8*(lane>>4)
    float2* o = &out[(size_t)(row0 + i + 8*hi)*NCC + lo];
    float2 v = *o; v.x += acc[i]; *o = v;
  }
}

// so3_linear on complex input (real weights): two WMMAs share B fragment
__global__ __launch_bounds__(32*WPB) void lin_cplx_kernel(
    const float2* __restrict__ X, const float* __restrict__ W,
    float2* __restrict__ out, int K) {
  int lane = threadIdx.x & 31, wave = threadIdx.x >> 5;
  int lo = lane & 15, hi = lane >> 4;
  int row0 = (blockIdx.x * WPB + wave) * 16;
  v8f cre = {}, cim = {};
  for (int k0 = 0; k0 < K; k0 += 4) {
    v2f ar, ai, b;
    #pragma unroll
    for (int j = 0; j < 2; ++j) {
      int kk = k0 + 2*hi + j;
      float2 A = X[(size_t)(row0 + lo)*K + kk];
      ar[j] = A.x; ai[j] = A.y;
      b[j]  = W[kk*NCC + lo];
    }
    cre = __builtin_amdgcn_wmma_f32_16x16x4_f32(false, ar, false, b, (short)0, cre,
                                                false, false);
    cim = __builtin_amdgcn_wmma_f32_16x16x4_f32(false, ai, false, b, (short)0, cim,
                                                false, false);
  }
  #pragma unroll
  for (int i = 0; i < 8; ++i) {
    float2* o = &out[(size_t)(row0 + i + 8*hi)*NCC + lo];
    float2 v = *o; v.x += cre[i]; v.y += cim[i]; *o = v;
  }
}

// Fused cg_product + so3_linear with async-to-LDS staged activation tiles.
//  out[n,m,d] += sum_pair sum_{c,cc} W[woff + c*ncb+cc, d] *
//                sum_terms C[a,b,m] * xa[n,a,c]*xb[n,b,cc]
// Per-wave LDS slab holds the 16-node xa/xb tiles (node stride padded by one
// float2 to break 64-bank alignment); A fragments synthesized from LDS.
__global__ __launch_bounds__(32*WPC) void cg_gemm_kernel(
    const float2* __restrict__ xa0, const float2* __restrict__ xa1,
    const float2* __restrict__ xa2, int nca,
    const float2* __restrict__ xb0, const float2* __restrict__ xb1,
    const float2* __restrict__ xb2, int ncb, int ncbShift,
    const float* __restrict__ W, float2* __restrict__ out, int outl,
    int npairs, int pt0, int pt1, int pt2,
    int maskA, int maskB, int sameAB, int ldsF2PerWave) {
  extern __shared__ float2 lds[];
  int lane = threadIdx.x & 31, wave = threadIdx.x >> 5;
  int lo = lane & 15, hi = lane >> 4;
  int node0 = (blockIdx.x * WPC + wave) * 16;
  int m = blockIdx.y;
  float2* slab = lds + (size_t)wave * ldsF2PerWave;

  // per-level slab offsets (padded node stride = (2l+1)*nc + 1 float2)
  int offA[3], offB[3];
  int run = 0;
  #pragma unroll
  for (int l = 0; l < 3; ++l) {
    offA[l] = run;
    if (maskA & (1 << l)) run += 16 * ((2*l + 1)*nca + 1);
  }
  #pragma unroll
  for (int l = 0; l < 3; ++l) {
    offB[l] = sameAB ? offA[l] : run;
    if (!sameAB && (maskB & (1 << l))) run += 16 * ((2*l + 1)*ncb + 1);
  }

  // stage activation tiles into this wave's slab
  for (int side = 0; side < (sameAB ? 1 : 2); ++side) {
    int msk = side ? maskB : maskA;
    int nc  = side ? ncb   : nca;
    for (int l = 0; l < 3; ++l) {
      if (!(msk & (1 << l))) continue;
      const float2* src = side ? ((l == 0) ? xb0 : (l == 1) ? xb1 : xb2)
                               : ((l == 0) ? xa0 : (l == 1) ? xa1 : xa2);
      int sa   = (2*l + 1) * nc;
      int base = side ? offB[l] : offA[l];
      for (int n = 0; n < 16; ++n) {
        const float2* g = src + (size_t)(node0 + n) * sa;
        float2*       d = slab + base + n * (sa + 1);
        for (int i = lane; i < sa; i += 32) stage_f2(g + i, d + i);
      }
    }
  }
  stage_wait();
  __syncthreads();

  v8f cre = {}, cim = {};
  int woff = 0;
  for (int p = 0; p < npairs; ++p) {
    int pt = (p == 0) ? pt0 : ((p == 1) ? pt1 : pt2);
    int la = PT_LA[pt], lb = PT_LB[pt];
    const float2* As = slab + offA[la];
    const float2* Bs = slab + offB[lb];
    int psa = (2*la + 1)*nca + 1;
    int psb = (2*lb + 1)*ncb + 1;
    int toff = CG_OFF[pt][m], tcnt = CG_CNT[pt][m];
    int K = nca * ncb;
    for (int k0 = 0; k0 < K; k0 += 4) {
      v2f ar, ai, b;
      #pragma unroll
      for (int j = 0; j < 2; ++j) {
        int kk = k0 + 2*hi + j;
        int c  = kk >> ncbShift;
        int cc = kk & (ncb - 1);
        float pr = 0.0f, pi = 0.0f;
        for (int t = 0; t < tcnt; ++t) {
          CGTerm tm = CG_TERMS[toff + t];
          float2 A = As[lo*psa + tm.a*nca + c];
          float2 B = Bs[lo*psb + tm.b*ncb + cc];
          pr += tm.c * (A.x*B.x - A.y*B.y);
          pi += tm.c * (A.x*B.y + A.y*B.x);
        }
        ar[j] = pr; ai[j] = pi;
        b[j]  = W[(size_t)(woff + kk)*NCC + lo];
      }
      cre = __builtin_amdgcn_wmma_f32_16x16x4_f32(false, ar, false, b, (short)0, cre,
                                                  false, false);
      cim = __builtin_amdgcn_wmma_f32_16x16x4_f32(false, ai, false, b, (short)0, cim,
                                                  false, false);
    }
    woff += K;
  }
  int stride = (2*outl + 1) * NCC;
  #pragma unroll
  for (int i = 0; i < 8; ++i) {
    float2* o = &out[(size_t)(node0 + i + 8*hi)*stride + m*NCC + lo];
    float2 v = *o; v.x += cre[i]; v.y += cim[i]; *o = v;
  }
}

// ---------------- readout --------------------------------------------------
__global__ __launch_bounds__(256) void readout_kernel(
    const float2* __restrict__ h2, const float* __restrict__ wf,
    const int* __restrict__ batch, float* __restrict__ sums,
    float* __restrict__ cnt) {
  int n = blockIdx.x * blockDim.x + threadIdx.x;
  if (n >= NN) return;
  float s = 0.0f;
  for (int d = 0; d < NCC; ++d) {
    float2 v = h2[(size_t)n*NCC + d];
    s += v.x * wf[d] + v.y * wf[NCC + d];
  }
  int g = batch[n];
  atomicAdd(&sums[g], s);
  atomicAdd(&cnt[g], 1.0f);
}

__global__ void finalize_kernel(const float* sums, const float* cnt, float* out) {
  int g = threadIdx.x;
  if (g < 8) out[g] = sums[g] / fmaxf(cnt[g], 1.0f);
}

// ---------------- host ------------------------------------------------------
extern "C" void kernel_launch(void* const* d_in, const int* in_sizes, int n_in,
                              void* d_out, int out_size, void* d_ws, size_t ws_size,
                              hipStream_t stream) {
  (void)in_sizes; (void)n_in; (void)out_size; (void)ws_size;
  // Inputs in setup_inputs() insertion order (params dict recursively):
  const float* x    = (const float*)d_in[0];   // (N,8)
  const float* pos  = (const float*)d_in[1];   // (N,3)
  const float* eatt = (const float*)d_in[2];   // (E,2)
  const int*   eidx = (const int*)d_in[3];     // (2,E)
  const int*   batch= (const int*)d_in[4];     // (N,)
  const float* wI0  = (const float*)d_in[5];   // w_init[0] (8,16)
  const float* wI1  = (const float*)d_in[6];   // w_init[1] (32,16)
  const float* b0xx[3] = {(const float*)d_in[7], (const float*)d_in[8], (const float*)d_in[9]};
  const float* b0yx[3] = {(const float*)d_in[10],(const float*)d_in[11],(const float*)d_in[12]};
  const float* b0yy[3] = {(const float*)d_in[13],(const float*)d_in[14],(const float*)d_in[15]};
  const float* b1xx = (const float*)d_in[16];
  const float* b1yx = (const float*)d_in[17];
  const float* b1yy = (const float*)d_in[18];
  const float* wfin = (const float*)d_in[19];  // (32,)

  char* ws = (char*)d_ws;
  size_t off = 0;
  auto alloc = [&](size_t nf2) -> float2* {
    float2* p = (float2*)(ws + off);
    off += ((nf2 * sizeof(float2)) + 255) & ~(size_t)255;
    return p;
  };
  float2* x1c  = alloc((size_t)NN*3*32);   // initial l=1 messages
  float2* h0_0 = alloc((size_t)NN*16);     // stage-0 x: l=0
  float2* h0_1 = alloc((size_t)NN*3*16);   //           l=1
  float2* y0_0 = alloc((size_t)NN*32);     // b0 mp:    l=0
  float2* y0_1 = alloc((size_t)NN*3*32);   //           l=1
  float2* h1_0 = alloc((size_t)NN*16);     // stage-1 x: l=0
  float2* h1_1 = alloc((size_t)NN*3*16);   //           l=1
  float2* h1_2 = alloc((size_t)NN*5*16);   //           l=2
  float2* y1_0 = alloc((size_t)NN*32);     // b1 mp:    l=0
  float2* y1_1 = alloc((size_t)NN*3*32);   //           l=1
  float2* y1_2 = alloc((size_t)NN*5*32);   //           l=2
  float2* h2_0 = alloc((size_t)NN*16);     // final l=0
  float* sums = (float*)(ws + off); off += 256;
  float* cnt  = (float*)(ws + off); off += 256;

  size_t totalFloats = off / 4;
  zero_kernel<<<(unsigned)((totalFloats + 255) / 256), 256, 0, stream>>>((float*)ws, totalFloats);

  // initial layer + so3_linear(w_init)
  init_edge_kernel<<<NE/256, 256, 0, stream>>>(pos, x, eidx, x1c);
  lin_real_kernel<<<(NN/16)/WPB,   32*WPB, 0, stream>>>(x,   wI0, h0_0, 8);
  lin_cplx_kernel<<<(NN*3/16)/WPB, 32*WPB, 0, stream>>>(x1c, wI1, h0_1, 32);

  // block 0 message passing
  mp_kernel<<<NE/256, 256, 0, stream>>>(h0_0, eidx, eatt, y0_0, 1, 16);
  mp_kernel<<<NE/256, 256, 0, stream>>>(h0_1, eidx, eatt, y0_1, 3, 16);

  static const int hLA[7] = {0,1,0,1,1,1,2};
  static const int hLB[7] = {0,1,1,0,1,1,2};
  auto cg = [&](const float2* a0, const float2* a1, const float2* a2, int nca,
                const float2* c0, const float2* c1, const float2* c2, int ncb,
                const float* W, float2* outp, int outl, int np, int p0, int p1, int p2) {
    int pts[3] = {p0, p1, p2};
    int maskA = 0, maskB = 0;
    for (int i = 0; i < np; ++i) { maskA |= 1 << hLA[pts[i]]; maskB |= 1 << hLB[pts[i]]; }
    int same = (a0 == c0 && a1 == c1 && a2 == c2 && nca == ncb) ? 1 : 0;
    if (same) { maskA |= maskB; maskB = maskA; }
    int f2 = 0;
    for (int l = 0; l < 3; ++l) if (maskA & (1 << l)) f2 += 16*((2*l+1)*nca + 1);
    if (!same)
      for (int l = 0; l < 3; ++l) if (maskB & (1 << l)) f2 += 16*((2*l+1)*ncb + 1);
    f2 = (f2 + 1) & ~1;                       // 16B-align wave slabs
    size_t ldsBytes = (size_t)f2 * sizeof(float2) * WPC;
    dim3 grid((NN/16)/WPC, 2*outl + 1);
    cg_gemm_kernel<<<grid, 32*WPC, ldsBytes, stream>>>(
        a0, a1, a2, nca, c0, c1, c2, ncb, (ncb == 32) ? 5 : 4,
        W, outp, outl, np, p0, p1, p2, maskA, maskB, same, f2);
  };

  // block 0: h1 = xx + yx + yy  (sequential accumulation on stream)
  cg(h0_0,h0_1,nullptr,16, h0_0,h0_1,nullptr,16, b0xx[0], h1_0, 0, 2, 0,1,0);
  cg(h0_0,h0_1,nullptr,16, h0_0,h0_1,nullptr,16, b0xx[1], h1_1, 1, 3, 2,3,4);
  cg(h0_0,h0_1,nullptr,16, h0_0,h0_1,nullptr,16, b0xx[2], h1_2, 2, 1, 5,0,0);
  cg(y0_0,y0_1,nullptr,32, h0_0,h0_1,nullptr,16, b0yx[0], h1_0, 0, 2, 0,1,0);
  cg(y0_0,y0_1,nullptr,32, h0_0,h0_1,nullptr,16, b0yx[1], h1_1, 1, 3, 2,3,4);
  cg(y0_0,y0_1,nullptr,32, h0_0,h0_1,nullptr,16, b0yx[2], h1_2, 2, 1, 5,0,0);
  cg(y0_0,y0_1,nullptr,32, y0_0,y0_1,nullptr,32, b0yy[0], h1_0, 0, 2, 0,1,0);
  cg(y0_0,y0_1,nullptr,32, y0_0,y0_1,nullptr,32, b0yy[1], h1_1, 1, 3, 2,3,4);
  cg(y0_0,y0_1,nullptr,32, y0_0,y0_1,nullptr,32, b0yy[2], h1_2, 2, 1, 5,0,0);

  // block 1 message passing
  mp_kernel<<<NE/256, 256, 0, stream>>>(h1_0, eidx, eatt, y1_0, 1, 16);
  mp_kernel<<<NE/256, 256, 0, stream>>>(h1_1, eidx, eatt, y1_1, 3, 16);
  mp_kernel<<<NE/256, 256, 0, stream>>>(h1_2, eidx, eatt, y1_2, 5, 16);

  // block 1: h2 = xx + yx + yy, out_ls=[0], pairs (0,0),(1,1),(2,2)
  cg(h1_0,h1_1,h1_2,16, h1_0,h1_1,h1_2,16, b1xx, h2_0, 0, 3, 0,1,6);
  cg(y1_0,y1_1,y1_2,32, h1_0,h1_1,h1_2,16, b1yx, h2_0, 0, 3, 0,1,6);
  cg(y1_0,y1_1,y1_2,32, y1_0,y1_1,y1_2,32, b1yy, h2_0, 0, 3, 0,1,6);

  // readout
  readout_kernel<<<NN/256, 256, 0, stream>>>(h2_0, wfin, batch, sums, cnt);
  finalize_kernel<<<1, 8, 0, stream>>>(sums, cnt, (float*)d_out);
}